// Proposal_48670569398568
// MI455X (gfx1250) — compile-verified
//
#include <hip/hip_runtime.h>
#include <stdint.h>

// ---------------- problem constants (match reference) ----------------
#define N_ANCH    4000000
#define MSEL      2000
#define NBINS     4096
#define CAND_CAP  8192        // padded candidate buffer (expected ~2.5K)
#define WORDS     64          // ceil(MSEL/32)=63, padded to 64
#define SCORE_THR 0.5f
#define NMS_THR   0.7f
#define IMG_WH    1024.0f
#define MIN_SZ    1.0f

// ---------------- workspace layout (bytes); needs ~650 KB ------------
constexpr size_t OFF_HIST  = 0;                      // 4096 * u32
constexpr size_t OFF_CTR   = OFF_HIST + NBINS * 4;   // [0]=cand_count [1]=thresh_bin [2]=count_above
constexpr size_t OFF_CAND  = 16512;                  // 8192 * u64 (16B aligned)
constexpr size_t OFF_TIDX  = OFF_CAND + CAND_CAP * 8;   // 2000 * i32
constexpr size_t OFF_VAL   = OFF_TIDX + MSEL * 4;       // 2000 * u32
constexpr size_t OFF_BOX   = OFF_VAL  + MSEL * 4;       // 2000 * float4 (16B aligned)
constexpr size_t OFF_KEEP0 = OFF_BOX  + MSEL * 16;      // 2000 * u32
constexpr size_t OFF_SUP   = OFF_KEEP0 + MSEL * 4;      // 2000 * 64 * u32 = 512 KB

// ---- CDNA5 async global->LDS path (guarded; fallback = sync copy) ----
#if defined(__has_builtin)
#if __has_builtin(__builtin_amdgcn_global_load_async_to_lds_b128) && \
    __has_builtin(__builtin_amdgcn_s_wait_asynccnt)
#define USE_ASYNC_LDS 1
#endif
#endif

#ifdef USE_ASYNC_LDS
// Exact pointee type demanded by the builtin (from hipcc diagnostic):
// '__attribute__((__vector_size__(4 * sizeof(int)))) int __device__ *'
typedef __attribute__((vector_size(4 * sizeof(int)))) int v4i_t;
typedef __attribute__((address_space(1))) v4i_t g_v4i;   // global (AS1)
typedef __attribute__((address_space(3))) v4i_t l_v4i;   // LDS (AS3)
#endif

// ---------------- K0: zero histogram + counters ----------------------
__global__ void k_init(uint32_t* hist, uint32_t* ctr) {
    int t = blockIdx.x * blockDim.x + threadIdx.x;
    if (t < NBINS) hist[t] = 0u;
    if (t < 4)     ctr[t]  = 0u;
}

// ---------------- K1: bucket histogram of fg scores -------------------
// score is (N,2); fg = score[:,1]. Read as float2 stream (32 MB, ~1.4us).
__global__ __launch_bounds__(256) void k_hist(const float2* __restrict__ score2,
                                              uint32_t* __restrict__ hist) {
    __shared__ uint32_t lh[NBINS];
    for (int i = threadIdx.x; i < NBINS; i += blockDim.x) lh[i] = 0u;
    __syncthreads();
    const int stride = gridDim.x * blockDim.x;
    for (int i = blockIdx.x * blockDim.x + threadIdx.x; i < N_ANCH; i += stride) {
        if (i + stride < N_ANCH) __builtin_prefetch(&score2[i + stride], 0, 0); // global_prefetch_b8
        float s = score2[i].y;
        if (s > SCORE_THR) {
            int b = (int)((s - SCORE_THR) * (2.0f * NBINS));
            b = min(max(b, 0), NBINS - 1);
            atomicAdd(&lh[b], 1u);
        }
    }
    __syncthreads();
    for (int i = threadIdx.x; i < NBINS; i += blockDim.x) {
        uint32_t c = lh[i];
        if (c) atomicAdd(&hist[i], c);
    }
}

// ---------------- K2: find threshold bin (suffix count from top) ------
__global__ void k_scan(const uint32_t* __restrict__ hist,
                       int* __restrict__ thresh_bin,
                       uint32_t* __restrict__ count_above) {
    __shared__ uint32_t chunk[256];   // each chunk = 16 bins, counted top-down
    int c = threadIdx.x;
    uint32_t s = 0;
    for (int k = 0; k < 16; ++k) s += hist[NBINS - 1 - (c * 16 + k)];
    chunk[c] = s;
    __syncthreads();
    if (threadIdx.x == 0) {
        uint32_t acc = 0; int bin = -1; uint32_t above = 0; int cc = 0;
        for (; cc < 256; ++cc) { if (acc + chunk[cc] >= MSEL) break; acc += chunk[cc]; }
        if (cc < 256) {
            for (int k = 0; k < 16; ++k) {
                int b = NBINS - 1 - (cc * 16 + k);
                uint32_t h = hist[b];
                if (acc + h >= MSEL) { bin = b; above = acc; break; }
                acc += h;
            }
        } else { bin = -1; above = acc; } // fewer than M candidates total: accept all
        *thresh_bin  = bin;
        *count_above = above;
    }
}

// ---------------- K3: compact candidates (second streaming pass) ------
// key = (score_bits << 32) | ~idx  -> sorting descending gives descending
// score with ascending index on ties, matching jax.lax.top_k order.
__global__ __launch_bounds__(256) void k_compact(const float2* __restrict__ score2,
                                                 const int* __restrict__ thresh_bin,
                                                 uint32_t* __restrict__ cand_count,
                                                 uint64_t* __restrict__ cand) {
    const int B = *thresh_bin;
    const int stride = gridDim.x * blockDim.x;
    for (int i = blockIdx.x * blockDim.x + threadIdx.x; i < N_ANCH; i += stride) {
        if (i + stride < N_ANCH) __builtin_prefetch(&score2[i + stride], 0, 0);
        float s = score2[i].y;
        if (s > SCORE_THR) {
            int b = min((int)((s - SCORE_THR) * (2.0f * NBINS)), NBINS - 1);
            if (b >= B) {
                uint32_t pos = atomicAdd(cand_count, 1u);
                if (pos < CAND_CAP) {
                    uint64_t key = ((uint64_t)__float_as_uint(s) << 32) |
                                   (uint64_t)(uint32_t)(~(uint32_t)i);
                    cand[pos] = key;
                }
            }
        }
    }
}

// ---------------- K4: one-block bitonic sort (desc) + take top M ------
// 8192 x u64 = 64 KB LDS (well under CDNA5's 320 KB/WGP).
__global__ __launch_bounds__(1024) void k_sort(const uint32_t* __restrict__ cand_count,
                                               const uint64_t* __restrict__ cand,
                                               int* __restrict__ top_idx,
                                               uint32_t* __restrict__ validf) {
    __shared__ uint64_t key[CAND_CAP];
    uint32_t cnt = *cand_count; if (cnt > CAND_CAP) cnt = CAND_CAP;
    for (int i = threadIdx.x; i < CAND_CAP; i += 1024)
        key[i] = (i < (int)cnt) ? cand[i] : 0ull;
    __syncthreads();
    for (int k = 2; k <= CAND_CAP; k <<= 1) {
        for (int j = k >> 1; j > 0; j >>= 1) {
            for (int i = threadIdx.x; i < CAND_CAP; i += 1024) {
                int p = i ^ j;
                if (p > i) {
                    uint64_t a = key[i], b = key[p];
                    bool dirDesc = ((i & k) == 0);
                    if ((a < b) == dirDesc) { key[i] = b; key[p] = a; }
                }
            }
            __syncthreads();
        }
    }
    for (int t = threadIdx.x; t < MSEL; t += 1024) {
        uint64_t kk = key[t];
        top_idx[t] = (kk != 0ull) ? (int)(~(uint32_t)kk) : 0;
        validf[t]  = (kk != 0ull) ? 1u : 0u;
    }
}

// ---------------- K5: gather + decode + clip + size filter ------------
__global__ __launch_bounds__(256) void k_decode(const float4* __restrict__ anchors4,
                                                const float4* __restrict__ deltas4,
                                                const int* __restrict__ top_idx,
                                                const uint32_t* __restrict__ validf,
                                                float4* __restrict__ boxes4,
                                                uint32_t* __restrict__ keep0) {
    int t = blockIdx.x * blockDim.x + threadIdx.x;
    if (t >= MSEL) return;
    int idx = top_idx[t];
    float4 a = anchors4[idx];
    float4 d = deltas4[idx];
    float w  = a.z - a.x, h = a.w - a.y;
    float cx = a.x + 0.5f * w, cy = a.y + 0.5f * h;
    float ncx = cx + d.x * w,  ncy = cy + d.y * h;
    float nw = w * expf(d.z),  nh = h * expf(d.w);
    float x0 = fminf(fmaxf(ncx - 0.5f * nw, 0.0f), IMG_WH);
    float y0 = fminf(fmaxf(ncy - 0.5f * nh, 0.0f), IMG_WH);
    float x1 = fminf(fmaxf(ncx + 0.5f * nw, 0.0f), IMG_WH);
    float y1 = fminf(fmaxf(ncy + 0.5f * nh, 0.0f), IMG_WH);
    bool big = ((x1 - x0) >= MIN_SZ) && ((y1 - y0) >= MIN_SZ);
    keep0[t]  = (validf[t] && big) ? 1u : 0u;
    boxes4[t] = make_float4(x0, y0, x1, y1);
}

// ---------------- K6: suppression bitmask via wave32 ballot -----------
// block = (32 lanes = bit index, 8 rows); 250 blocks cover 2000 rows.
// Boxes (32 KB) staged in LDS via CDNA5 async global->LDS when available.
__global__ __launch_bounds__(256) void k_mask(const float4* __restrict__ boxes4,
                                              uint32_t* __restrict__ supmask) {
    __shared__ float4 sb[MSEL];
    const int tid = threadIdx.y * 32 + threadIdx.x;
#ifdef USE_ASYNC_LDS
    for (int k = tid; k < MSEL; k += 256) {
        __builtin_amdgcn_global_load_async_to_lds_b128(
            (g_v4i*)&boxes4[k],          // AS1 int4* (const cast away; builtin is non-const)
            (l_v4i*)&sb[k],              // AS3 int4*
            0, 0);
    }
    __builtin_amdgcn_s_wait_asynccnt(0);
    __syncthreads();
#else
    for (int k = tid; k < MSEL; k += 256) sb[k] = boxes4[k];
    __syncthreads();
#endif
    const int i    = blockIdx.x * 8 + threadIdx.y;   // row, always < MSEL
    const int lane = threadIdx.x;
    float4 bi = sb[i];
    float areai = (bi.z - bi.x) * (bi.w - bi.y);
    for (int w = 0; w < WORDS; ++w) {
        int j = w * 32 + lane;
        bool sup = false;
        if (j < MSEL && j > i) {
            float4 bj = sb[j];
            float areaj = (bj.z - bj.x) * (bj.w - bj.y);
            float lx = fmaxf(bi.x, bj.x), ly = fmaxf(bi.y, bj.y);
            float rx = fminf(bi.z, bj.z), ry = fminf(bi.w, bj.w);
            float iw = fmaxf(rx - lx, 0.0f), ih = fmaxf(ry - ly, 0.0f);
            float inter = iw * ih;
            float iou = inter / (areai + areaj - inter + 1e-9f);
            sup = iou > NMS_THR;
        }
        uint32_t m = __builtin_amdgcn_ballot_w32(sup);   // wave32 mask word
        if (lane == 0) supmask[i * WORDS + w] = m;
    }
}

// ---------------- K7: sequential greedy reduce + write outputs --------
// Single 64-thread block (2 waves); removed-state + keep flags in LDS.
__global__ __launch_bounds__(64) void k_nms_out(const uint32_t* __restrict__ keep0,
                                                const uint32_t* __restrict__ supmask,
                                                const float4* __restrict__ boxes4,
                                                float* __restrict__ out) {
    __shared__ uint32_t removed[WORDS];
    __shared__ uint32_t kf[MSEL];
    const int tid = threadIdx.x;
    removed[tid] = 0u;
    for (int t = tid; t < MSEL; t += 64) kf[t] = keep0[t];
    __syncthreads();
    for (int i = 0; i < MSEL; ++i) {
        uint32_t row = supmask[i * WORDS + tid];          // prefetch row (unconditional)
        bool k = kf[i] && !((removed[i >> 5] >> (i & 31)) & 1u);
        __syncthreads();                                  // reads done before writes
        if (k) removed[tid] |= row;
        if (tid == 0) kf[i] = k ? 1u : 0u;                // finalize keep[i] in place
        __syncthreads();                                  // writes visible to next iter
    }
    // outputs: boxes_out (2000x4) then keep (2000) as float
    for (int t = tid; t < MSEL; t += 64) {
        bool k = kf[t] != 0u;
        float4 b = boxes4[t];
        out[t * 4 + 0] = k ? b.x : 0.0f;
        out[t * 4 + 1] = k ? b.y : 0.0f;
        out[t * 4 + 2] = k ? b.z : 0.0f;
        out[t * 4 + 3] = k ? b.w : 0.0f;
        out[MSEL * 4 + t] = k ? 1.0f : 0.0f;
    }
}

// ---------------------------------------------------------------------
extern "C" void kernel_launch(void* const* d_in, const int* in_sizes, int n_in,
                              void* d_out, int out_size, void* d_ws, size_t ws_size,
                              hipStream_t stream) {
    (void)in_sizes; (void)n_in; (void)out_size; (void)ws_size;
    const float4* anchors = (const float4*)d_in[0];   // (N,4) f32
    const float2* score   = (const float2*)d_in[1];   // (N,2) f32
    const float4* deltas  = (const float4*)d_in[2];   // (N,4) f32
    char* ws = (char*)d_ws;
    uint32_t* hist  = (uint32_t*)(ws + OFF_HIST);
    uint32_t* ctr   = (uint32_t*)(ws + OFF_CTR);
    uint64_t* cand  = (uint64_t*)(ws + OFF_CAND);
    int*      tidx  = (int*)(ws + OFF_TIDX);
    uint32_t* val   = (uint32_t*)(ws + OFF_VAL);
    float4*   boxes = (float4*)(ws + OFF_BOX);
    uint32_t* keep0 = (uint32_t*)(ws + OFF_KEEP0);
    uint32_t* sup   = (uint32_t*)(ws + OFF_SUP);
    float*    out   = (float*)d_out;

    k_init   <<<16, 256, 0, stream>>>(hist, ctr);
    k_hist   <<<512, 256, 0, stream>>>(score, hist);
    k_scan   <<<1, 256, 0, stream>>>(hist, (int*)(ctr + 1), ctr + 2);
    k_compact<<<512, 256, 0, stream>>>(score, (const int*)(ctr + 1), ctr + 0, cand);
    k_sort   <<<1, 1024, 0, stream>>>(ctr + 0, cand, tidx, val);
    k_decode <<<8, 256, 0, stream>>>(anchors, deltas, tidx, val, boxes, keep0);
    k_mask   <<<dim3(250), dim3(32, 8), 0, stream>>>(boxes, sup);
    k_nms_out<<<1, 64, 0, stream>>>(keep0, sup, boxes, out);
}